// BlockLinear_27384711479625
// MI455X (gfx1250) — compile-verified
//
#include <hip/hip_runtime.h>

// Block-diagonal GEMM: for b in 0..15: out[b*256:(b+1)*256, :] = W[b] @ inp[b*256:(b+1)*256, :]
// fp32 end-to-end via V_WMMA_F32_16X16X4_F32 (exact precision match with the fp32 reference).
// Async global->LDS (ASYNCcnt) double-buffered pipeline; each workgroup covers the full
// M=256 of one diagonal block so X is streamed from HBM exactly once (~272 MB total).

#define NBLK   16
#define BS     256      // square block size == K == workgroup M tile
#define NCOLS  8192
#define NT     64       // workgroup N tile
#define KT     32       // K chunk per pipeline stage
#define LDA    36       // KT + 4 pad (floats)  -> conflict-free A fragment reads
#define LDB    72       // NT + 8 pad (floats)  -> conflict-free B fragment reads

typedef __attribute__((ext_vector_type(2))) float v2f;
typedef __attribute__((ext_vector_type(8))) float v8f;
typedef __attribute__((address_space(3))) const float lds_cf;

__device__ __forceinline__ unsigned to_lds_off(const void* p) {
    // generic -> LDS address space: numeric value is the LDS byte offset
    return (unsigned)(unsigned long long)(lds_cf*)(const float*)p;
}

__device__ __forceinline__ void async_b128(unsigned lds_byte, const void* gbase, unsigned gbyte) {
    // GVS mode: mem = SADDR64 + VADDR32; VDST vgpr carries the LDS byte address.
    asm volatile("global_load_async_to_lds_b128 %0, %1, %2 offset:0"
                 :
                 : "v"(lds_byte), "v"(gbyte), "s"(gbase)
                 : "memory");
}

__device__ __forceinline__ void wait_async_all() {
    asm volatile("s_wait_asynccnt 0" ::: "memory");
}

__global__ __launch_bounds__(256) void BlockLinear_27384711479625_kernel(
    const float* __restrict__ inp, const float* __restrict__ W, float* __restrict__ out) {
    __shared__ __align__(16) float As[2][BS * LDA];   // W tile: 256 x KT (padded)   72 KB
    __shared__ __align__(16) float Bs[2][KT * LDB];   // X tile: KT x NT  (padded)   18 KB

    const int t    = threadIdx.x;
    const int lane = t & 31;
    const int wv   = t >> 5;        // wave 0..7 -> 32-row M strip
    const int m    = lane & 15;
    const int half = lane >> 4;     // K-split / M-split lane half per ISA fp32 layout

    const int n0 = blockIdx.x * NT;
    const int bb = blockIdx.y;

    const float* Wb = W   + (size_t)bb * BS * BS;     // 256x256 block, row-major
    const float* Xb = inp + (size_t)bb * BS * NCOLS;  // 256x8192 row panel

    const unsigned aB0 = to_lds_off(&As[0][0]);
    const unsigned aB1 = to_lds_off(&As[1][0]);
    const unsigned bB0 = to_lds_off(&Bs[0][0]);
    const unsigned bB1 = to_lds_off(&Bs[1][0]);

    auto load_chunk = [&](int kc, int buf) {
        const int k0 = kc * KT;
        const unsigned aB = buf ? aB1 : aB0;
        const unsigned bB = buf ? bB1 : bB0;
        // A: BS*KT floats = 2048 float4; 8 per thread
#pragma unroll
        for (int i = 0; i < 8; ++i) {
            const int q  = i * 256 + t;
            const int r  = q >> 3;       // row 0..255 (8 float4 per row)
            const int c4 = q & 7;
            async_b128(aB + (unsigned)((r * LDA + c4 * 4) * 4), Wb,
                       (unsigned)((r * BS + k0 + c4 * 4) * 4));
        }
        // B: KT*NT floats = 512 float4; 2 per thread
#pragma unroll
        for (int i = 0; i < 2; ++i) {
            const int q  = i * 256 + t;
            const int r  = q >> 4;       // row 0..31 (16 float4 per row)
            const int c4 = q & 15;
            async_b128(bB + (unsigned)((r * LDB + c4 * 4) * 4), Xb,
                       (unsigned)(((k0 + r) * NCOLS + n0 + c4 * 4) * 4));
        }
    };

    // 32(M) x 64(N) wave tile: 2 A-fragments x 4 B-fragments -> 8 WMMA per k-step
    v8f c00 = {}, c01 = {}, c02 = {}, c03 = {};
    v8f c10 = {}, c11 = {}, c12 = {}, c13 = {};
    const int arow0 = wv * 32 + m;
    const int arow1 = arow0 + 16;

    load_chunk(0, 0);

    const int NCHUNK = BS / KT;   // 8
#pragma unroll 1
    for (int kc = 0; kc < NCHUNK; ++kc) {
        const int buf = kc & 1;
        wait_async_all();
        __syncthreads();
        if (kc + 1 < NCHUNK) load_chunk(kc + 1, (kc + 1) & 1);

        const float* Ab = &As[buf][0];
        const float* Bb = &Bs[buf][0];
#pragma unroll
        for (int k = 0; k < KT; k += 4) {
            const int kk = k + 2 * half;          // ISA A/B fp32 lane-half K split
            v2f a0 = *(const v2f*)&Ab[arow0 * LDA + kk];
            v2f a1 = *(const v2f*)&Ab[arow1 * LDA + kk];
            v2f b0, b1, b2, b3;
            b0.x = Bb[kk * LDB +  0 + m]; b0.y = Bb[(kk + 1) * LDB +  0 + m];
            b1.x = Bb[kk * LDB + 16 + m]; b1.y = Bb[(kk + 1) * LDB + 16 + m];
            b2.x = Bb[kk * LDB + 32 + m]; b2.y = Bb[(kk + 1) * LDB + 32 + m];
            b3.x = Bb[kk * LDB + 48 + m]; b3.y = Bb[(kk + 1) * LDB + 48 + m];
            c00 = __builtin_amdgcn_wmma_f32_16x16x4_f32(false, a0, false, b0, (short)0, c00, false, false);
            c01 = __builtin_amdgcn_wmma_f32_16x16x4_f32(false, a0, false, b1, (short)0, c01, false, false);
            c02 = __builtin_amdgcn_wmma_f32_16x16x4_f32(false, a0, false, b2, (short)0, c02, false, false);
            c03 = __builtin_amdgcn_wmma_f32_16x16x4_f32(false, a0, false, b3, (short)0, c03, false, false);
            c10 = __builtin_amdgcn_wmma_f32_16x16x4_f32(false, a1, false, b0, (short)0, c10, false, false);
            c11 = __builtin_amdgcn_wmma_f32_16x16x4_f32(false, a1, false, b1, (short)0, c11, false, false);
            c12 = __builtin_amdgcn_wmma_f32_16x16x4_f32(false, a1, false, b2, (short)0, c12, false, false);
            c13 = __builtin_amdgcn_wmma_f32_16x16x4_f32(false, a1, false, b3, (short)0, c13, false, false);
        }
    }

    // C/D layout: VGPR r -> M = r (lanes 0-15) / r+8 (lanes 16-31); N = lane%16
    {
        const int row0 = bb * BS + wv * 32 + half * 8;        // mi = 0 strip
        float* o0 = out + (size_t)row0 * NCOLS + (n0 + m);
        float* o1 = o0 + (size_t)16 * NCOLS;                  // mi = 1 strip (+16 rows)
#pragma unroll
        for (int r = 0; r < 8; ++r) {
            o0[(size_t)r * NCOLS +  0] = c00[r];
            o0[(size_t)r * NCOLS + 16] = c01[r];
            o0[(size_t)r * NCOLS + 32] = c02[r];
            o0[(size_t)r * NCOLS + 48] = c03[r];
            o1[(size_t)r * NCOLS +  0] = c10[r];
            o1[(size_t)r * NCOLS + 16] = c11[r];
            o1[(size_t)r * NCOLS + 32] = c12[r];
            o1[(size_t)r * NCOLS + 48] = c13[r];
        }
    }
}

extern "C" void kernel_launch(void* const* d_in, const int* in_sizes, int n_in,
                              void* d_out, int out_size, void* d_ws, size_t ws_size,
                              hipStream_t stream) {
    (void)in_sizes; (void)n_in; (void)out_size; (void)d_ws; (void)ws_size;
    const float* inp = (const float*)d_in[0];
    const float* W   = (const float*)d_in[1];
    float* out       = (float*)d_out;
    dim3 grid(NCOLS / NT, NBLK, 1);   // 128 x 16
    BlockLinear_27384711479625_kernel<<<grid, dim3(256, 1, 1), 0, stream>>>(inp, W, out);
}